// GAT_28836410425899
// MI455X (gfx1250) — compile-verified
//
#include <hip/hip_runtime.h>
#include <hip/hip_bf16.h>

#define NFEAT 256
#define NHID  64
#define NHEAD 8
#define NOUT  128

typedef __attribute__((ext_vector_type(2))) float v2f;
typedef __attribute__((ext_vector_type(8))) float v8f;

// ---------------------------------------------------------------------------
// Dense fp32 GEMM on the matrix pipe: C[M,N] = A[M,K] * B[K,N], row-major.
// Each wave computes an MT x NT block of 16x16 C tiles (register blocking:
// A fragment reused NT times, B fragment reused MT times -> 1.5 DWORDs of
// load per V_WMMA_F32_16X16X4_F32 at MT=4,NT=2). Software-pipelined: the
// k+4 fragments are fetched into a second register set while the k-step's
// 8 WMMAs issue, so VMEM latency overlaps the matrix pipe instead of
// serializing on s_wait_loadcnt 0.
// fp32 in / fp32 accumulate: numerically matches the fp32 JAX reference.
//
// Fragment layouts (ISA 7.12.2):
//  A 16x4 : lanes 0-15 (M=lane) K={k,k+1}; lanes 16-31 (M=lane-16) K={k+2,k+3}
//           -> one 8-byte v2f load per lane per k-step.
//  C 16x16: VGPR r -> M=r (lanes 0-15) / M=8+r (lanes 16-31), N = lane&15.
// ---------------------------------------------------------------------------
template <int MT, int NT>
__global__ void __launch_bounds__(256)
gat_wmma_gemm_f32(const float* __restrict__ A, const float* __restrict__ B,
                  float* __restrict__ C, int M, int K, int N) {
  const int lane    = threadIdx.x & 31;
  const int ngroups = N / (NT * 16);
  const int mgroups = (M + MT * 16 - 1) / (MT * 16);
  const int g = blockIdx.x * (blockDim.x >> 5) + (threadIdx.x >> 5);
  if (g >= mgroups * ngroups) return;
  const int mg = g / ngroups;
  const int ng = g - mg * ngroups;

  const int half = lane >> 4;   // 0: K pair {0,1}, 1: K pair {2,3}
  const int l    = lane & 15;

  const float* ap[MT];
#pragma unroll
  for (int i = 0; i < MT; ++i) {
    int row = mg * (MT * 16) + i * 16 + l;
    if (row >= M) row = M - 1;              // clamp reads; stores predicated below
    ap[i] = A + (size_t)row * K + half * 2;
    __builtin_prefetch(ap[i], 0, 3);        // global_prefetch_b8: warm WGP$ for A row
  }
  const float* bp[NT];
#pragma unroll
  for (int j = 0; j < NT; ++j)
    bp[j] = B + (size_t)(half * 2) * N + ng * (NT * 16) + j * 16 + l;

  v8f acc[MT][NT] = {};

  // --- software pipeline: prologue loads k=0 fragments ---
  v2f a0[MT], b0[NT];
#pragma unroll
  for (int i = 0; i < MT; ++i) a0[i] = *(const v2f*)(ap[i]);
#pragma unroll
  for (int j = 0; j < NT; ++j) { b0[j].x = bp[j][0]; b0[j].y = bp[j][N]; }

#pragma unroll 2
  for (int k = 4; k < K; k += 4) {
    // fetch next k-step into the shadow register set (overlaps the WMMAs)
    v2f a1[MT], b1[NT];
#pragma unroll
    for (int i = 0; i < MT; ++i) a1[i] = *(const v2f*)(ap[i] + k);
#pragma unroll
    for (int j = 0; j < NT; ++j) {
      b1[j].x = bp[j][(size_t)k * N];       // row k + 2*half
      b1[j].y = bp[j][(size_t)k * N + N];   // row k + 2*half + 1
    }
    // consume current fragments: 8 WMMAs, no same-iteration load dependency
#pragma unroll
    for (int i = 0; i < MT; ++i)
#pragma unroll
      for (int j = 0; j < NT; ++j)
        acc[i][j] = __builtin_amdgcn_wmma_f32_16x16x4_f32(
            false, a0[i], false, b0[j], (short)0, acc[i][j], false, false);
    // rotate
#pragma unroll
    for (int i = 0; i < MT; ++i) a0[i] = a1[i];
#pragma unroll
    for (int j = 0; j < NT; ++j) b0[j] = b1[j];
  }
  // epilogue: last k-step
#pragma unroll
  for (int i = 0; i < MT; ++i)
#pragma unroll
    for (int j = 0; j < NT; ++j)
      acc[i][j] = __builtin_amdgcn_wmma_f32_16x16x4_f32(
          false, a0[i], false, b0[j], (short)0, acc[i][j], false, false);

#pragma unroll
  for (int i = 0; i < MT; ++i) {
#pragma unroll
    for (int j = 0; j < NT; ++j) {
      const int col = ng * (NT * 16) + j * 16 + l;
#pragma unroll
      for (int r = 0; r < 8; ++r) {
        int row = mg * (MT * 16) + i * 16 + half * 8 + r;
        if (row < M) C[(size_t)row * N + col] = acc[i][j][r];
      }
    }
  }
}

// ---------------------------------------------------------------------------
// el[n,h] = dot(feat[n,h,:], al[h,:]); er likewise. One wave32 per (n,h).
// ---------------------------------------------------------------------------
template <int H, int D>
__global__ void __launch_bounds__(256)
gat_el_er(const float* __restrict__ feat, const float* __restrict__ al,
          const float* __restrict__ ar, float* __restrict__ el,
          float* __restrict__ er, int n) {
  const int wg   = blockIdx.x * (blockDim.x >> 5) + (threadIdx.x >> 5);
  const int lane = threadIdx.x & 31;
  if (wg >= n * H) return;
  const int h = wg % H;                      // H is a power-of-two constant
  const float* f = feat + (size_t)wg * D;    // feat is [n,H,D]; wg == n*H + h
  const float* a = al + (size_t)h * D;
  const float* r = ar + (size_t)h * D;
  float sl = 0.f, sr = 0.f;
#pragma unroll
  for (int d = lane; d < D; d += 32) {
    float v = f[d];
    sl += v * a[d];
    sr += v * r[d];
  }
#pragma unroll
  for (int off = 16; off > 0; off >>= 1) {
    sl += __shfl_xor(sl, off, 32);
    sr += __shfl_xor(sr, off, 32);
  }
  if (lane == 0) { el[wg] = sl; er[wg] = sr; }
}

// monotonic float<->int map so float max can use native integer atomicMax
__device__ __forceinline__ int fmono(float v) {
  int i = __float_as_int(v);
  return i >= 0 ? i : (i ^ 0x7fffffff);
}
__device__ __forceinline__ float fmono_dec(int i) {
  return __int_as_float(i >= 0 ? i : (i ^ 0x7fffffff));
}

__device__ __forceinline__ void edge_sd(const int* __restrict__ ei, int E,
                                        int e, int& s, int& d) {
  if (e < E) { s = ei[e]; d = ei[E + e]; }
  else       { s = d = e - E; }              // appended self-loops
}

template <int H>
__global__ void gat_edge_max(const int* __restrict__ ei, int E, int Etot,
                             const float* __restrict__ el,
                             const float* __restrict__ er,
                             int* __restrict__ m) {
  int t = blockIdx.x * blockDim.x + threadIdx.x;
  if (t >= Etot * H) return;
  int e = t / H, h = t - e * H;
  int s, d; edge_sd(ei, E, e, s, d);
  float v = el[s * H + h] + er[d * H + h];
  v = v >= 0.f ? v : 0.2f * v;               // leaky_relu(0.2)
  atomicMax(m + d * H + h, fmono(v));
}

template <int H>
__global__ void gat_edge_exp(const int* __restrict__ ei, int E, int Etot,
                             const float* __restrict__ el,
                             const float* __restrict__ er,
                             const int* __restrict__ m,
                             float* __restrict__ ex, float* __restrict__ den) {
  int t = blockIdx.x * blockDim.x + threadIdx.x;
  if (t >= Etot * H) return;
  int e = t / H, h = t - e * H;
  int s, d; edge_sd(ei, E, e, s, d);
  float v = el[s * H + h] + er[d * H + h];
  v = v >= 0.f ? v : 0.2f * v;
  float x = __expf(v - fmono_dec(m[d * H + h]));
  ex[(size_t)e * H + h] = x;
  atomicAdd(den + d * H + h, x);
}

// rst[dst,h,:] += (ex/den) * feat[src,h,:]; one thread per 4 features per edge
template <int H, int D>
__global__ void gat_edge_agg(const int* __restrict__ ei, int E, int Etot,
                             const float* __restrict__ feat,
                             const float* __restrict__ ex,
                             const float* __restrict__ den,
                             float* __restrict__ rst) {
  constexpr int F = H * D;
  constexpr int CH = F >> 2;                 // float4 chunks per edge
  long long t = (long long)blockIdx.x * blockDim.x + threadIdx.x;
  if (t >= (long long)Etot * CH) return;
  int e = (int)(t / CH);
  int c = (int)(t - (long long)e * CH);
  int s, d; edge_sd(ei, E, e, s, d);
  int h = (c << 2) / D;                      // constant shift
  float a = ex[(size_t)e * H + h] / den[(size_t)d * H + h];
  const float4 f = *(const float4*)(feat + (size_t)s * F + (c << 2));
  float* o = rst + (size_t)d * F + (c << 2);
  atomicAdd(o + 0, a * f.x);
  atomicAdd(o + 1, a * f.y);
  atomicAdd(o + 2, a * f.z);
  atomicAdd(o + 3, a * f.w);
}

__global__ void gat_elu_bias(const float* __restrict__ rst,
                             const float* __restrict__ bias,
                             float* __restrict__ out, size_t n, int F) {
  size_t i = (size_t)blockIdx.x * blockDim.x + threadIdx.x;
  if (i >= n) return;
  float v = rst[i] + bias[i & (size_t)(F - 1)];   // F is 512 (pow2)
  out[i] = v > 0.f ? v : (__expf(v) - 1.f);       // ELU
}

__global__ void gat_add_bias(const float* __restrict__ rst,
                             const float* __restrict__ bias,
                             float* __restrict__ out, size_t n, int F) {
  size_t i = (size_t)blockIdx.x * blockDim.x + threadIdx.x;
  if (i >= n) return;
  out[i] = rst[i] + bias[i & (size_t)(F - 1)];    // F is 128 (pow2)
}

__global__ void gat_fill_f32(float* __restrict__ p, float v, size_t n) {
  size_t i = (size_t)blockIdx.x * blockDim.x + threadIdx.x;
  if (i < n) p[i] = v;
}
__global__ void gat_fill_i32(int* __restrict__ p, int v, size_t n) {
  size_t i = (size_t)blockIdx.x * blockDim.x + threadIdx.x;
  if (i < n) p[i] = v;
}

static inline int ceil_div_ll(long long a, int b) { return (int)((a + b - 1) / b); }

extern "C" void kernel_launch(void* const* d_in, const int* in_sizes, int n_in,
                              void* d_out, int out_size, void* d_ws, size_t ws_size,
                              hipStream_t stream) {
  const float* x   = (const float*)d_in[0];
  const int*   ei  = (const int*)d_in[1];   // jnp int64 request -> int32 under default JAX x64-off
  const float* W1  = (const float*)d_in[2];
  const float* al1 = (const float*)d_in[3];
  const float* ar1 = (const float*)d_in[4];
  const float* b1  = (const float*)d_in[5];
  const float* W2  = (const float*)d_in[6];
  const float* al2 = (const float*)d_in[7];
  const float* ar2 = (const float*)d_in[8];
  const float* b2  = (const float*)d_in[9];

  const int n    = in_sizes[0] / NFEAT;     // 50000
  const int E    = in_sizes[1] / 2;         // 400000
  const int Etot = E + n;                   // + self loops
  const int F1   = NHEAD * NHID;            // 512
  const int F2   = NOUT;                    // 128

  // workspace layout (floats)
  float* bufA = (float*)d_ws;                       // n*512 : feat1, later h=elu(...)
  float* bufB = bufA + (size_t)n * F1;              // n*512 : rst1, later feat2|rst2
  float* ex   = bufB + (size_t)n * F1;              // Etot*8 (layer1) / Etot (layer2)
  float* el   = ex + (size_t)Etot * NHEAD;          // n*8
  float* er   = el + (size_t)n * NHEAD;             // n*8
  int*   mb   = (int*)(er + (size_t)n * NHEAD);     // n*8
  float* den  = (float*)(mb + (size_t)n * NHEAD);   // n*8

  const int B256 = 256;
  constexpr int MT = 4, NT = 2;             // 64x32 C block per wave

  // ---------------- layer 1 (H=8, D=64) ----------------
  {
    // feat1 = x @ W1   [n,256]x[256,512]
    long long waves = (long long)((n + MT * 16 - 1) / (MT * 16)) * (F1 / (NT * 16));
    gat_wmma_gemm_f32<MT, NT><<<ceil_div_ll(waves, 8), B256, 0, stream>>>(
        x, W1, bufA, n, NFEAT, F1);
    gat_el_er<NHEAD, NHID><<<ceil_div_ll((long long)n * NHEAD, 8), B256, 0, stream>>>(
        bufA, al1, ar1, el, er, n);
    gat_fill_i32<<<ceil_div_ll((long long)n * NHEAD, B256), B256, 0, stream>>>(
        mb, (int)0x80000000, (size_t)n * NHEAD);
    gat_fill_f32<<<ceil_div_ll((long long)n * NHEAD, B256), B256, 0, stream>>>(
        den, 0.f, (size_t)n * NHEAD);
    gat_edge_max<NHEAD><<<ceil_div_ll((long long)Etot * NHEAD, B256), B256, 0, stream>>>(
        ei, E, Etot, el, er, mb);
    gat_edge_exp<NHEAD><<<ceil_div_ll((long long)Etot * NHEAD, B256), B256, 0, stream>>>(
        ei, E, Etot, el, er, mb, ex, den);
    gat_fill_f32<<<ceil_div_ll((long long)n * F1, B256), B256, 0, stream>>>(
        bufB, 0.f, (size_t)n * F1);
    gat_edge_agg<NHEAD, NHID><<<ceil_div_ll((long long)Etot * (F1 >> 2), B256), B256, 0, stream>>>(
        ei, E, Etot, bufA, ex, den, bufB);
    gat_elu_bias<<<ceil_div_ll((long long)n * F1, B256), B256, 0, stream>>>(
        bufB, b1, bufA, (size_t)n * F1, F1);
  }

  // ---------------- layer 2 (H=1, D=128) ----------------
  {
    float* feat2 = bufB;                     // n*128
    float* rst2  = bufB + (size_t)n * F2;    // n*128 (within bufB's n*512)
    long long waves = (long long)((n + MT * 16 - 1) / (MT * 16)) * (F2 / (NT * 16));
    gat_wmma_gemm_f32<MT, NT><<<ceil_div_ll(waves, 8), B256, 0, stream>>>(
        bufA, W2, feat2, n, F1, F2);
    gat_el_er<1, NOUT><<<ceil_div_ll((long long)n, 8), B256, 0, stream>>>(
        feat2, al2, ar2, el, er, n);
    gat_fill_i32<<<ceil_div_ll((long long)n, B256), B256, 0, stream>>>(
        mb, (int)0x80000000, (size_t)n);
    gat_fill_f32<<<ceil_div_ll((long long)n, B256), B256, 0, stream>>>(
        den, 0.f, (size_t)n);
    gat_edge_max<1><<<ceil_div_ll((long long)Etot, B256), B256, 0, stream>>>(
        ei, E, Etot, el, er, mb);
    gat_edge_exp<1><<<ceil_div_ll((long long)Etot, B256), B256, 0, stream>>>(
        ei, E, Etot, el, er, mb, ex, den);
    gat_fill_f32<<<ceil_div_ll((long long)n * F2, B256), B256, 0, stream>>>(
        rst2, 0.f, (size_t)n * F2);
    gat_edge_agg<1, NOUT><<<ceil_div_ll((long long)Etot * (F2 >> 2), B256), B256, 0, stream>>>(
        ei, E, Etot, feat2, ex, den, rst2);
    gat_add_bias<<<ceil_div_ll((long long)n * F2, B256), B256, 0, stream>>>(
        rst2, b2, (float*)d_out, (size_t)n * F2, F2);
  }
}